// CenterNetLoss_58317065945825
// MI455X (gfx1250) — compile-verified
//
#include <hip/hip_runtime.h>
#include <math.h>

#define B_ 16
#define C_ 80
#define H_ 128
#define W_ 128
#define N_ 64
#define HW_ (H_*W_)
#define STRIDE_ 4.0f
#define ALPHA_ 0.54f
#define EPS_ 1.1920929e-07f

typedef float v2f __attribute__((ext_vector_type(2)));
typedef float v4f __attribute__((ext_vector_type(4)));
typedef float v8f __attribute__((ext_vector_type(8)));

// workspace layout (floats)
#define NBLK1    2048
#define OFF_BASE 0
#define OFF_CORR (OFF_BASE + NBLK1)      // 2048 .. +1024
#define OFF_POS  (OFF_CORR + B_*N_)      // 3072 .. +1024
#define OFF_BOXA (OFF_POS  + B_*N_)      // 4096 .. +16
#define OFF_BOXC (OFF_BOXA + B_)         // 4112 .. +16

__device__ __forceinline__ float sigmoid_(float x) { return 1.0f / (1.0f + expf(-x)); }

// baseline focal term for target==0:  -log(clip(1-p,1e-12)) * p^2
__device__ __forceinline__ float neg0_(float x) {
    float p = sigmoid_(x);
    float q = 1.0f - p;
    return -logf(fmaxf(q, 1e-12f)) * p * p;
}

// Deterministic 256-thread block reduction.
// Stage 1: LDS fold 256 -> 32 (fixed order).
// Stage 2: V_WMMA_F32_16X16X4_F32 with A = ones(16x4): D[m][n] = sum_k B[k][n].
//          Lane L<16 holds B[0][L]=a_L (b1=0); lane L>=16 holds B[2][L-16]=a_L.
//          => D[0][n] = a_n + a_{n+16}: folds 32 -> 16 on the matrix pipe.
// Stage 3: thread 0 sums 16 values in fixed order. Result valid on thread 0.
__device__ float block_reduce_sum_wmma(float v, float* s) {
    const int t = threadIdx.x;
    s[t] = v;
    __syncthreads();
    if (t < 32) {                        // wave 0 only: EXEC all ones within wave
        float a = 0.0f;
#pragma unroll
        for (int i = 0; i < 8; ++i) a += s[t + 32 * i];
        v2f av; av.x = 1.0f; av.y = 1.0f;   // A = ones (16x4)
        v2f bv; bv.x = a;    bv.y = 0.0f;   // B rows 0/2 carry data, rows 1/3 zero
        v8f c = {};
        c = __builtin_amdgcn_wmma_f32_16x16x4_f32(
                false, av, false, bv, (short)0, c, false, false);
        if (t < 16) s[t] = c[0];            // D[0][n], lanes 0..15
    }
    __syncthreads();
    float r = 0.0f;
    if (t == 0) {
#pragma unroll
        for (int i = 0; i < 16; ++i) r += s[i];
    }
    __syncthreads();                        // protect LDS for back-to-back calls
    return r;
}

// ---------------- Kernel 1: streaming baseline focal (memory bound) --------
__global__ void k_focal_base(const float* __restrict__ hm, float* __restrict__ ws) {
    __shared__ float sred[256];
    const v4f* hm4 = (const v4f*)hm;
    const int total4 = (B_ * C_ * H_ * W_) / 4;   // 5,242,880
    const int stride = gridDim.x * blockDim.x;
    float acc = 0.0f;
    for (int i = blockIdx.x * blockDim.x + threadIdx.x; i < total4; i += stride) {
        if (i + stride < total4)
            __builtin_prefetch(hm4 + i + stride, 0, 1);   // global_prefetch_b8
        v4f v = __builtin_nontemporal_load(hm4 + i);      // streamed once: NT b128
        acc += neg0_(v[0]) + neg0_(v[1]) + neg0_(v[2]) + neg0_(v[3]);
    }
    float r = block_reduce_sum_wmma(acc, sred);
    if (threadIdx.x == 0) ws[OFF_BASE + blockIdx.x] = r;
}

// ---------------- Kernel 2: sparse gaussian-window corrections -------------
// One block per (box n, image b). Window offsets in [-7,8]^2 cover rmax=7.
__global__ void k_correction(const float* __restrict__ hm,
                             const float* __restrict__ boxes,
                             const int*   __restrict__ labels,
                             float* __restrict__ ws) {
    __shared__ float scx[N_], scy[N_], srx[N_], sry[N_], sinvx[N_], sinvy[N_];
    __shared__ int   slbl[N_];
    __shared__ float sred[256];
    const int b = blockIdx.y;
    const int n = blockIdx.x;
    const int t = threadIdx.x;

    if (t < N_) {
        const float* bp = boxes + (b * N_ + t) * 4;
        float x = bp[0] / STRIDE_, y = bp[1] / STRIDE_;
        float w = bp[2] / STRIDE_, h = bp[3] / STRIDE_;
        float cx = rintf(x + w * 0.5f);
        float cy = rintf(y + h * 0.5f);
        float rx = fmaxf(0.0f, rintf(w * 0.5f * ALPHA_));
        float ry = fmaxf(0.0f, rintf(h * 0.5f * ALPHA_));
        float sx = rx / 3.0f + (1.0f / 6.0f);
        float sy = ry / 3.0f + (1.0f / 6.0f);
        scx[t] = cx; scy[t] = cy; srx[t] = rx; sry[t] = ry;
        sinvx[t] = 1.0f / (2.0f * sx * sx);
        sinvy[t] = 1.0f / (2.0f * sy * sy);
        slbl[t] = labels[b * N_ + t];
    }
    __syncthreads();

    const int tx = t & 15, ty = t >> 4;
    const float dxo = (float)(tx - 7);
    const float dyo = (float)(ty - 7);
    float corr = 0.0f, pos = 0.0f;

    if (fabsf(dxo) <= srx[n] && fabsf(dyo) <= sry[n]) {
        float px = scx[n] + dxo, py = scy[n] + dyo;
        if (px >= 0.0f && px <= (float)(W_ - 1) && py >= 0.0f && py <= (float)(H_ - 1)) {
            const int myl = slbl[n];
            float gmax = 0.0f; int owner = -1;
            for (int m = 0; m < N_; ++m) {            // per-class max (dedup rule)
                if (slbl[m] != myl) continue;
                float dx = px - scx[m], dy = py - scy[m];
                if (fabsf(dx) <= srx[m] && fabsf(dy) <= sry[m]) {
                    float gx = expf(-(dx * dx) * sinvx[m]);
                    float gy = expf(-(dy * dy) * sinvy[m]);
                    float g = gy * gx;
                    if (g < EPS_) g = 0.0f;
                    if (g > gmax) { gmax = g; owner = m; }   // first argmax owns
                }
            }
            if (owner == n) {                          // exactly one block corrects
                int xi = (int)px, yi = (int)py;
                float logit = hm[((b * C_ + myl) * H_ + yi) * W_ + xi];
                float p = sigmoid_(logit);
                float base = -logf(fmaxf(1.0f - p, 1e-12f)) * p * p;
                if (gmax == 1.0f) {                    // exact center: pos pixel
                    float omp = 1.0f - p;
                    corr = -logf(fmaxf(p, 1e-12f)) * omp * omp - base;
                    pos = 1.0f;
                } else {                               // penalty-reduced neg term
                    float u = 1.0f - gmax;
                    float u4 = u * u; u4 = u4 * u4;
                    corr = base * (u4 - 1.0f);
                }
            }
        }
    }
    float rc = block_reduce_sum_wmma(corr, sred);
    if (t == 0) ws[OFF_CORR + b * N_ + n] = rc;
    float rp = block_reduce_sum_wmma(pos, sred);
    if (t == 0) ws[OFF_POS + b * N_ + n] = rp;
}

// ---------------- Kernel 3: 3x3 decoded box L1 loss ------------------------
__global__ void k_box(const float* __restrict__ bo, const float* __restrict__ boxes,
                      float* __restrict__ ws) {
    __shared__ float sred[256];
    const int b = blockIdx.x;
    const int t = threadIdx.x;
    float aacc = 0.0f, vacc = 0.0f;
    for (int s = t; s < N_ * 9; s += 256) {
        int n = s / 9, k = s - n * 9;
        int di = k / 3 - 1;          // x offset (axis 2 in reference)
        int dj = k % 3 - 1;          // y offset (axis 3)
        const float* bp = boxes + (b * N_ + n) * 4;
        float bx = bp[0], by = bp[1], bw = bp[2], bh = bp[3];
        float x = bx / STRIDE_, y = by / STRIDE_, w = bw / STRIDE_, h = bh / STRIDE_;
        int cxi = (int)rintf(x + w * 0.5f);
        int cyi = (int)rintf(y + h * 0.5f);
        int gx = cxi + di, gy = cyi + dj;
        bool valid = (gx >= 0) && (gx <= W_ - 1) && (gy >= 0) && (gy <= H_ - 1);
        if (valid) {
            int idx = gy * W_ + gx;
            float o0 = bo[(b * 4 + 0) * HW_ + idx];
            float o1 = bo[(b * 4 + 1) * HW_ + idx];
            float o2 = bo[(b * 4 + 2) * HW_ + idx];
            float o3 = bo[(b * 4 + 3) * HW_ + idx];
            float fx = (float)gx, fy = (float)gy;
            aacc += fabsf((fx - o0) * STRIDE_ - bx)
                  + fabsf((fy - o1) * STRIDE_ - by)
                  + fabsf((fx + o2) * STRIDE_ - (bx + bw))
                  + fabsf((fy + o3) * STRIDE_ - (by + bh));
            vacc += 1.0f;
        }
    }
    float ra = block_reduce_sum_wmma(aacc, sred);
    if (t == 0) ws[OFF_BOXA + b] = ra;
    float rv = block_reduce_sum_wmma(vacc, sred);
    if (t == 0) ws[OFF_BOXC + b] = rv;
}

// ---------------- Kernel 4: deterministic final combine --------------------
__global__ void k_final(const float* __restrict__ ws, float* __restrict__ out) {
    __shared__ float sred[256];
    const int t = threadIdx.x;
    float v = 0.0f;
#pragma unroll
    for (int i = 0; i < NBLK1 / 256; ++i) v += ws[OFF_BASE + t + 256 * i];
#pragma unroll
    for (int i = 0; i < (B_ * N_) / 256; ++i) v += ws[OFF_CORR + t + 256 * i];
    float heat = block_reduce_sum_wmma(v, sred);

    float vp = 0.0f;
#pragma unroll
    for (int i = 0; i < (B_ * N_) / 256; ++i) vp += ws[OFF_POS + t + 256 * i];
    float npos = block_reduce_sum_wmma(vp, sred);

    if (t == 0) {
        float bsum = 0.0f;
        for (int b = 0; b < B_; ++b) {
            float cnt4 = ws[OFF_BOXC + b] * 4.0f;
            bsum += ws[OFF_BOXA + b] / fmaxf(cnt4, 1.0f);
        }
        out[0] = heat / fmaxf(npos, 1.0f);   // heatmap (focal) loss
        out[1] = bsum / (float)B_;           // box loss
    }
}

extern "C" void kernel_launch(void* const* d_in, const int* in_sizes, int n_in,
                              void* d_out, int out_size, void* d_ws, size_t ws_size,
                              hipStream_t stream) {
    const float* hm     = (const float*)d_in[0];   // heatmap     [B,C,H,W] f32
    const float* bo     = (const float*)d_in[1];   // box_offsets [B,4,H,W] f32
    const float* boxes  = (const float*)d_in[2];   // boxes       [B,N,4]   f32
    const int*   labels = (const int*)  d_in[3];   // labels      [B,N]     i32
    float* ws  = (float*)d_ws;
    float* out = (float*)d_out;

    k_focal_base<<<NBLK1, 256, 0, stream>>>(hm, ws);
    k_correction<<<dim3(N_, B_), 256, 0, stream>>>(hm, boxes, labels, ws);
    k_box<<<B_, 256, 0, stream>>>(bo, boxes, ws);
    k_final<<<1, 256, 0, stream>>>(ws, out);
}